// DensityCalculator_61727269978252
// MI455X (gfx1250) — compile-verified
//
#include <hip/hip_runtime.h>

// CDNA5 / gfx1250 density map kernel.
// density[g] = sum_a sum_n (aw[a,n]*mask[a]) * exp(bw[a,n] * max(||g - x_a||^2, 0))
//
// Strategy: V_WMMA_F32_16X16X4_F32 computes 16x16 dist^2 tiles
//   (A = [grid|1] 16x4, B = [-2X | ||x||^2] 4x16, C = ||g||^2 rows),
// TRANS pipe (v_exp_f32) + FMAs do the Gaussian accumulation,
// LDS holds all atom tables (staged once per workgroup).

typedef float v2f __attribute__((ext_vector_type(2)));
typedef float v8f __attribute__((ext_vector_type(8)));

#define GPTS   (48 * 48 * 48)   // 110592 grid points
#define NATOMS 800
#define NTILES (NATOMS / 16)    // 50 atom tiles
#define NGAUSS 6
#define LOG2E  1.4426950408889634f

__device__ __forceinline__ float fast_exp2(float x) {
#if __has_builtin(__builtin_amdgcn_exp2f)
  return __builtin_amdgcn_exp2f(x);   // raw v_exp_f32
#else
  return exp2f(x);
#endif
}

__global__ __launch_bounds__(256) void density_wmma_kernel(
    const float* __restrict__ X,          // [800,3]
    const float* __restrict__ aw,         // [800,6]
    const float* __restrict__ bw,         // [800,6]
    const int*   __restrict__ elements,   // [800]
    const int*   __restrict__ cexpand,    // [800]
    const float* __restrict__ real_grid,  // [G,3]
    float*       __restrict__ out)        // [G]
{
  // LDS atom tables: 12800 + 25600 + 25600 = 64000 bytes (<< 320KB/WGP)
  __shared__ float sB [NATOMS * 4];   // per atom: {-2x, -2y, -2z, |x|^2}  (B operand)
  __shared__ float sBW[NATOMS * 8];   // bw * log2(e), padded to stride 8
  __shared__ float sW [NATOMS * 8];   // aw * mask,    padded to stride 8

  for (int a = threadIdx.x; a < NATOMS; a += blockDim.x) {
    float x = X[a * 3 + 0], y = X[a * 3 + 1], z = X[a * 3 + 2];
    sB[a * 4 + 0] = -2.0f * x;
    sB[a * 4 + 1] = -2.0f * y;
    sB[a * 4 + 2] = -2.0f * z;
    sB[a * 4 + 3] = x * x + y * y + z * z;
    float m = (elements[a] != 5 && cexpand[a] == 1) ? 1.0f : 0.0f;
#pragma unroll
    for (int j = 0; j < NGAUSS; ++j) {
      sBW[a * 8 + j] = bw[a * 6 + j] * LOG2E;
      sW [a * 8 + j] = aw[a * 6 + j] * m;
    }
  }
  __syncthreads();

  const int lane  = threadIdx.x & 31;     // wave32
  const int wave  = threadIdx.x >> 5;     // 8 waves / block
  const int half  = lane >> 4;            // 0: lanes 0-15, 1: lanes 16-31
  const int n     = lane & 15;            // column (atom-in-tile) index
  const int k0    = 2 * half;             // K components this half-wave carries
  const int gBase = (blockIdx.x * 8 + wave) * 16;   // 16 grid points per wave

  // --- A fragment (loop invariant): row M = n, components K = k0, k0+1 ---
  // f32 16x4 A layout: VGPR j, lane half h -> K = 2h + j.  A[m,3] = 1.0.
  v2f afrag;
  {
    const float* rg = real_grid + (size_t)(gBase + n) * 3;
    float c0 = rg[0], c1 = rg[1], c2 = rg[2];
    afrag[0] = half ? c2   : c0;
    afrag[1] = half ? 1.0f : c1;
  }

  // --- C fragment (loop invariant): C[m,*] = ||g_m||^2, row m = r + 8*half ---
  v8f cfrag;
#pragma unroll
  for (int r = 0; r < 8; ++r) {
    const float* rg = real_grid + (size_t)(gBase + 8 * half + r) * 3;
    cfrag[r] = rg[0] * rg[0] + rg[1] * rg[1] + rg[2] * rg[2];
  }

  float acc[8];
#pragma unroll
  for (int r = 0; r < 8; ++r) acc[r] = 0.0f;

  // --- main loop over 50 atom tiles ---
  for (int t = 0; t < NTILES; ++t) {
    const int a = t * 16 + n;   // this lane's atom column

    // B fragment: B[K = k0+j][n] = sB[a].{component k0+j}  (one ds_load_b64)
    const float* bp = &sB[a * 4 + k0];
    v2f bfrag;
    bfrag[0] = bp[0];
    bfrag[1] = bp[1];

    // D = A x B + C  ->  16x16 tile of squared distances
    v8f d = __builtin_amdgcn_wmma_f32_16x16x4_f32(
        /*neg_a=*/false, afrag, /*neg_b=*/false, bfrag,
        /*c_mod=*/(short)0, cfrag, /*reuse_a=*/false, /*reuse_b=*/false);

    // Gaussian tables for this lane's atom (shared across its 8 rows)
    float bwl[NGAUSS], wl[NGAUSS];
#pragma unroll
    for (int j = 0; j < NGAUSS; ++j) {
      bwl[j] = sBW[a * 8 + j];
      wl[j]  = sW [a * 8 + j];
    }

    // 48 exp + 48 FMA per lane per tile (TRANS-pipe bound, as intended)
#pragma unroll
    for (int r = 0; r < 8; ++r) {
      float dd = fmaxf(d[r], 0.0f);
      float s = 0.0f;
#pragma unroll
      for (int j = 0; j < NGAUSS; ++j)
        s = fmaf(wl[j], fast_exp2(bwl[j] * dd), s);
      acc[r] += s;
    }
  }

  // --- reduce each row across the 16 column lanes (stays inside half-wave) ---
#pragma unroll
  for (int r = 0; r < 8; ++r) {
    float v = acc[r];
    v += __shfl_xor(v, 1, 32);
    v += __shfl_xor(v, 2, 32);
    v += __shfl_xor(v, 4, 32);
    v += __shfl_xor(v, 8, 32);
    acc[r] = v;
  }

  // lane with n == r writes row r of its half (rows 0-7 lower, 8-15 upper)
#pragma unroll
  for (int r = 0; r < 8; ++r)
    if (n == r) out[gBase + 8 * half + r] = acc[r];
}

extern "C" void kernel_launch(void* const* d_in, const int* in_sizes, int n_in,
                              void* d_out, int out_size, void* d_ws, size_t ws_size,
                              hipStream_t stream) {
  (void)in_sizes; (void)n_in; (void)d_ws; (void)ws_size; (void)out_size;
  const float* X         = (const float*)d_in[0];
  const float* aw        = (const float*)d_in[1];
  const float* bw        = (const float*)d_in[2];
  const int*   elements  = (const int*)  d_in[3];
  const int*   cexpand   = (const int*)  d_in[4];
  const float* real_grid = (const float*)d_in[5];
  float*       out       = (float*)d_out;

  // 16 grid points per wave, 8 waves per block -> 128 points per block
  dim3 grid(GPTS / 128);   // 864 blocks
  dim3 block(256);
  density_wmma_kernel<<<grid, block, 0, stream>>>(
      X, aw, bw, elements, cexpand, real_grid, out);
}